// CNN_12970801234173
// MI455X (gfx1250) — compile-verified
//
#include <hip/hip_runtime.h>

// ---------------------------------------------------------------------------
// SpiralNet encoder: 4 spiral-gather GEMM layers + final linear, all fp32.
// Hout[m, o] = sum_f G[m, f] * W[o, f] + bias[o],
//   G[m, j*CIN + c] = Hin[b, idx[n, j], c],  m = b*NV + n.
// Matrix pipe: V_WMMA_F32_16X16X4_F32.
// Staging: GLOBAL_LOAD_ASYNC_TO_LDS_B128 (per-lane gather straight to LDS,
// ASYNCcnt-tracked), double-buffered LDS software pipeline.
// ---------------------------------------------------------------------------

typedef __attribute__((ext_vector_type(2))) float v2f;
typedef __attribute__((ext_vector_type(8))) float v8f;

constexpr int NV    = 5023;
constexpr int BATCH = 32;
constexpr int MTOT  = NV * BATCH;   // 160736 GEMM rows
constexpr int BM    = 128;          // rows per block (8 waves x 16 rows)

// Low 32 bits of a generic pointer to LDS == LDS byte address (aperture rule).
__device__ __forceinline__ unsigned lds_addr32(const void* p) {
    return (unsigned)(unsigned long long)p;
}

// Per-lane async gather: 16B global -> LDS, tracked by ASYNCcnt.
__device__ __forceinline__ void async_load_b128(unsigned lds, const float* g) {
    asm volatile("global_load_async_to_lds_b128 %0, %1, off"
                 :: "v"(lds), "v"(g) : "memory");
}

template <int N>
__device__ __forceinline__ void wait_async() {
#if __has_builtin(__builtin_amdgcn_s_wait_asynccnt)
    __builtin_amdgcn_s_wait_asynccnt(N);
#else
    asm volatile("s_wait_asynccnt %0" :: "i"(N) : "memory");
#endif
}

template <int CIN, int COUT, int SPIRAL, int NT, bool GATHER>
__global__ __launch_bounds__(256)
void spiral_gemm(const float* __restrict__ Hin,   // [BATCH, NV, CIN]
                 const int*   __restrict__ idx,   // [NV, SPIRAL] (this layer)
                 const float* __restrict__ Wt,    // [COUT, SPIRAL*CIN]
                 const float* __restrict__ bias,  // [COUT]
                 float*       __restrict__ Hout)  // [BATCH, NV, COUT]
{
    constexpr bool VEC = (CIN % 4) == 0;               // async path usable
    constexpr int KC   = VEC ? 32 : ((CIN + 3) & ~3);  // K-chunk
    constexpr int LDW  = KC + 4;          // LDS row stride (16B align, bank-skewed)
    constexpr int BN   = NT * 16;         // N columns per block
    constexpr int FAN  = SPIRAL * CIN;    // W row stride
    constexpr int CPS  = (CIN + KC - 1) / KC;   // K-chunks per spiral step
    constexpr int NS   = SPIRAL * CPS;          // total pipeline stages
    constexpr int KC4  = KC / 4;
    constexpr int SGE  = BM * LDW;        // floats per G buffer
    constexpr int SWE  = BN * LDW;        // floats per W buffer
    constexpr int GOPS = (BM * KC4) / 256;      // async ops/thread, G stage
    constexpr int WOPS = (BN * KC4) / 256;      // async ops/thread, W stage
    constexpr int WAITN = GOPS + WOPS;          // outstanding after next issue
    constexpr int GOPSA = (GOPS > 0) ? GOPS : 1;  // array dims (>=1)
    constexpr int WOPSA = (WOPS > 0) ? WOPS : 1;

    __shared__ float sG[2 * SGE];         // double-buffered gather tile
    __shared__ float sW[2 * SWE];         // double-buffered weight tile

    const int tid  = threadIdx.x;
    const int wave = tid >> 5;
    const int lane = tid & 31;
    const int lrow = lane & 15;           // N (B/C/D frags) or M (A frag)
    const int khi  = (lane >> 4) << 1;    // K offset 0 / 2 per half-wave

    const int m0 = blockIdx.x * BM;
    const int n0 = blockIdx.y * BN;
    const bool full = (m0 + BM) <= MTOT;  // all 128 rows valid -> async path

    // --- accumulators initialized with bias (C/D lane -> N = lrow) ----------
    v8f acc[NT];
#pragma unroll
    for (int t = 0; t < NT; ++t) {
        const float bv = bias[n0 + t * 16 + lrow];
        v8f a;
#pragma unroll
        for (int r = 0; r < 8; ++r) a[r] = bv;
        acc[t] = a;
    }

    const int arow = wave * 16 + lrow;    // this wave's A-fragment row

    // --- stage-invariant gather descriptors (hoisted out of the pipeline) ---
    unsigned gBase[GOPSA];   // b*NV*CIN + 4*c4  (element offset)
    int      gIdx[GOPSA];    // GATHER ? n*SPIRAL : n
    unsigned gLoff[GOPSA];   // LDS byte offset within G buffer
    unsigned wBase[WOPSA];
    unsigned wLoff[WOPSA];
    const unsigned sGbase = lds_addr32(sG);
    const unsigned sWbase = lds_addr32(sW);
    if constexpr (VEC) {
        // G: thread handles elements i = tid + k*256 -> (row r, 16B column c4)
#pragma unroll
        for (int k = 0; k < GOPS; ++k) {
            const int i  = tid + k * 256;
            const int r  = i / KC4;
            const int c4 = i - r * KC4;
            const int m  = m0 + r;
            const int mm = (m < MTOT) ? m : 0;   // full-path never uses clamped
            const unsigned b = (unsigned)mm / NV;
            const int n = mm - (int)b * NV;
            gBase[k] = b * (unsigned)(NV * CIN) + 4u * c4;
            gIdx[k]  = GATHER ? n * SPIRAL : n;
            gLoff[k] = (unsigned)((r * LDW + 4 * c4) * 4);
        }
        // W: thread handles elements i = tid + k*256 -> (out chan o, column c4)
#pragma unroll
        for (int k = 0; k < WOPS; ++k) {
            const int i  = tid + k * 256;
            const int o  = i / KC4;
            const int c4 = i - o * KC4;
            wBase[k] = (unsigned)(n0 + o) * FAN + 4u * c4;
            wLoff[k] = (unsigned)((o * LDW + 4 * c4) * 4);
        }
    }

    // ---- stage issue: fill parity buffer with chunk s ----------------------
    auto issue_stage = [&](int s, int par) {
        const int j  = s / CPS;
        const int c0 = (s - j * CPS) * KC;

        if constexpr (VEC) {
            if (full) {
                const unsigned gB = sGbase + (unsigned)(par * SGE * 4);
                const unsigned wB = sWbase + (unsigned)(par * SWE * 4);
#pragma unroll
                for (int k = 0; k < GOPS; ++k) {
                    int src;
                    if constexpr (GATHER) src = idx[gIdx[k] + j];
                    else                  src = gIdx[k];
                    async_load_b128(gB + gLoff[k],
                                    Hin + (size_t)(gBase[k] + (unsigned)src * CIN + c0));
                }
#pragma unroll
                for (int k = 0; k < WOPS; ++k) {
                    async_load_b128(wB + wLoff[k],
                                    Wt + (size_t)(wBase[k] + (unsigned)(j * CIN + c0)));
                }
                return;
            }
        }

        // synchronous fallback: partial last M-block, or CIN==3 (zero-padded)
        float* gb = sG + par * SGE;
        float* wb = sW + par * SWE;
        for (int i = tid; i < BM * KC; i += 256) {
            const int r  = i / KC;
            const int c  = i - r * KC;
            const int m  = m0 + r;
            const int cc = c0 + c;
            float val = 0.f;
            if (m < MTOT && cc < CIN) {
                const unsigned b = (unsigned)m / NV;
                const int n = m - (int)b * NV;
                int src;
                if constexpr (GATHER) src = idx[n * SPIRAL + j];
                else                  src = n;
                val = Hin[((size_t)b * NV + src) * CIN + cc];
            }
            gb[r * LDW + c] = val;
        }
        for (int i = tid; i < BN * KC; i += 256) {
            const int o  = i / KC;
            const int c  = i - o * KC;
            const int cc = c0 + c;
            wb[o * LDW + c] = (cc < CIN)
                ? Wt[(size_t)(n0 + o) * FAN + j * CIN + cc] : 0.f;
        }
    };

    // ---- software pipeline: issue(s+1) | wait(s) | compute(s) --------------
    issue_stage(0, 0);
    for (int s = 0; s < NS; ++s) {
        const int par = s & 1;
        if (s + 1 < NS) {
            issue_stage(s + 1, par ^ 1);
            wait_async<WAITN>();   // async loads retire in order -> stage s landed
        } else {
            wait_async<0>();
        }
        __syncthreads();           // stage-s LDS data visible to all waves

        const float* gb = sG + par * SGE;
        const float* wb = sW + par * SWE;
#pragma unroll
        for (int k0 = 0; k0 < KC; k0 += 4) {
            v2f afrag;   // A: lane = M row, VGPR r -> K = k0 + khi + r
            afrag.x = gb[arow * LDW + k0 + khi];
            afrag.y = gb[arow * LDW + k0 + khi + 1];
#pragma unroll
            for (int t = 0; t < NT; ++t) {
                v2f bfrag;  // B: lane = N col, same K mapping
                bfrag.x = wb[(t * 16 + lrow) * LDW + k0 + khi];
                bfrag.y = wb[(t * 16 + lrow) * LDW + k0 + khi + 1];
                acc[t] = __builtin_amdgcn_wmma_f32_16x16x4_f32(
                    false, afrag, false, bfrag, (short)0, acc[t], false, false);
            }
        }
        __syncthreads();           // done reading buf[par] before reuse
    }

    // ---- store: D VGPR r -> M = r + 8*(lane>=16), lane -> N ----------------
    const int mbase = m0 + wave * 16 + (lane >> 4) * 8;
#pragma unroll
    for (int t = 0; t < NT; ++t) {
        const int ncol = n0 + t * 16 + lrow;
#pragma unroll
        for (int r = 0; r < 8; ++r) {
            const int m = mbase + r;
            if (m < MTOT)
                Hout[(size_t)m * COUT + ncol] = acc[t][r];
        }
    }
}

// ---------------------------------------------------------------------------
// Launch: 5 GEMM layers, activations ping-pong through workspace.
// Inputs: x, spiral_idx[4,NV,9], W0,b0, W1,b1, W2,b2, W3,b3, Wl,bl
// ---------------------------------------------------------------------------
extern "C" void kernel_launch(void* const* d_in, const int* in_sizes, int n_in,
                              void* d_out, int out_size, void* d_ws, size_t ws_size,
                              hipStream_t stream) {
    const float* x  = (const float*)d_in[0];
    const int*   sp = (const int*)  d_in[1];
    const float* W0 = (const float*)d_in[2];
    const float* b0 = (const float*)d_in[3];
    const float* W1 = (const float*)d_in[4];
    const float* b1 = (const float*)d_in[5];
    const float* W2 = (const float*)d_in[6];
    const float* b2 = (const float*)d_in[7];
    const float* W3 = (const float*)d_in[8];
    const float* b3 = (const float*)d_in[9];
    const float* Wl = (const float*)d_in[10];
    const float* bl = (const float*)d_in[11];
    float* out = (float*)d_out;

    float* bufA = (float*)d_ws;                       // h1 (32c), h3 (128c)
    float* bufB = bufA + (size_t)MTOT * 128;          // h2 (64c), h4 (128c)

    const dim3 blk(256);
    const int  gx = (MTOT + BM - 1) / BM;             // 1256 M-tiles
    const int  S9 = NV * 9;                           // spiral_idx layer stride

    spiral_gemm<  3,  32, 9, 2, true ><<<dim3(gx, 1), blk, 0, stream>>>(x,    sp,        W0, b0, bufA);
    spiral_gemm< 32,  64, 9, 4, true ><<<dim3(gx, 1), blk, 0, stream>>>(bufA, sp + S9,   W1, b1, bufB);
    spiral_gemm< 64, 128, 9, 8, true ><<<dim3(gx, 1), blk, 0, stream>>>(bufB, sp + 2*S9, W2, b2, bufA);
    spiral_gemm<128, 128, 9, 8, true ><<<dim3(gx, 1), blk, 0, stream>>>(bufA, sp + 3*S9, W3, b3, bufB);
    spiral_gemm<128, 256, 1, 8, false><<<dim3(gx, 2), blk, 0, stream>>>(bufB, nullptr,   Wl, bl, out);
}